// Model_25262997636024
// MI455X (gfx1250) — compile-verified
//
#include <hip/hip_runtime.h>

// y = x * x[0]; output is (y, y) concatenated: d_out[0..n) and d_out[n..2n).
// Streaming kernel: per thread, 4x B128 NT loads then 8x B128 NT stores.
// HBM-bound: 384 MB total traffic -> ~16.5 us floor at 23.3 TB/s.

typedef float v4f __attribute__((ext_vector_type(4)));  // native clang vector: OK for nontemporal builtins

constexpr int TPB = 256;  // 8 wave32s per block
constexpr int VPT = 4;    // float4 granules per thread

__global__ __launch_bounds__(TPB) void scale_dup_v4(const float* __restrict__ x,
                                                    float* __restrict__ out,
                                                    int n) {
    // Uniform address -> scalar load (s_load_b32), broadcast to the wave.
    const float s = x[0];

    const int nvec = n >> 2;  // n % 4 == 0 guaranteed by host-side dispatch
    const int base = blockIdx.x * (TPB * VPT) + threadIdx.x;

    const v4f* xv = (const v4f*)x;
    v4f* o0 = (v4f*)out;
    v4f* o1 = (v4f*)(out + n);  // 16B-aligned since n % 4 == 0

    if (base + (VPT - 1) * TPB < nvec) {
        // Fast path (all full blocks): no predication, loads form one clause
        // with 4 outstanding global_load_b128, wave-contiguous 512B segments.
        v4f v[VPT];
#pragma unroll
        for (int j = 0; j < VPT; ++j) {
            v[j] = __builtin_nontemporal_load(xv + base + j * TPB);
        }
        // NT stores: the 256MB write stream must not thrash the 192MB L2
        // (every byte is touched exactly once).
#pragma unroll
        for (int j = 0; j < VPT; ++j) {
            const v4f r = v[j] * s;
            __builtin_nontemporal_store(r, o0 + base + j * TPB);
            __builtin_nontemporal_store(r, o1 + base + j * TPB);
        }
    } else {
        // Edge path: last (partial) block only.
#pragma unroll
        for (int j = 0; j < VPT; ++j) {
            const int idx = base + j * TPB;
            if (idx < nvec) {
                const v4f r = __builtin_nontemporal_load(xv + idx) * s;
                __builtin_nontemporal_store(r, o0 + idx);
                __builtin_nontemporal_store(r, o1 + idx);
            }
        }
    }
}

// General fallback for n % 4 != 0 (not taken for n = 32M).
__global__ __launch_bounds__(TPB) void scale_dup_scalar(const float* __restrict__ x,
                                                        float* __restrict__ out,
                                                        int n) {
    const float s = x[0];
    const int i = blockIdx.x * blockDim.x + threadIdx.x;
    if (i < n) {
        const float r = x[i] * s;
        __builtin_nontemporal_store(r, out + i);
        __builtin_nontemporal_store(r, out + n + i);
    }
}

extern "C" void kernel_launch(void* const* d_in, const int* in_sizes, int n_in,
                              void* d_out, int out_size, void* d_ws, size_t ws_size,
                              hipStream_t stream) {
    const float* x = (const float*)d_in[0];
    float* out = (float*)d_out;
    const int n = in_sizes[0];

    if ((n & 3) == 0) {
        const int nvec = n >> 2;
        const int per_block = TPB * VPT;  // 1024 float4s = 4096 floats per block
        const int blocks = (nvec + per_block - 1) / per_block;
        scale_dup_v4<<<blocks, TPB, 0, stream>>>(x, out, n);
    } else {
        const int blocks = (n + TPB - 1) / TPB;
        scale_dup_scalar<<<blocks, TPB, 0, stream>>>(x, out, n);
    }
}